// Database_52931176956568
// MI455X (gfx1250) — compile-verified
//
#include <hip/hip_runtime.h>
#include <math.h>

// ---------------------------------------------------------------------------
// MI455X (gfx1250) fused L1-normalize -> f32 WMMA GEMM -> masked top-4
//   TDM (tensor_load_to_lds) double-buffered streaming of embedding slabs.
//   query [256,768] f32, embeddings [768,400000] f32
//   out: values[256,4] then indices[256,4] (as float), flat.
// ---------------------------------------------------------------------------

typedef float v2f __attribute__((ext_vector_type(2)));
typedef float v4f __attribute__((ext_vector_type(4)));
typedef float v8f __attribute__((ext_vector_type(8)));
typedef unsigned int u32x4 __attribute__((ext_vector_type(4)));
typedef int i32x4 __attribute__((ext_vector_type(4)));
typedef int i32x8 __attribute__((ext_vector_type(8)));

#define Q_DIM   256
#define D_DIM   768
#define NTILE   64
#define KTILE   32          // K-slab rows per TDM transfer (24 slabs per chunk)
#define BSTRIDE 72          // 64 + 8 pad dwords (TDM pad: interval=64dw, amount=8dw)
#define SSTRIDE 34          // 32 + 2 pad: per-row scan hits 32 distinct banks per wave
#define GRID_MAIN 512

// ---------------------------------------------------------------------------
// Kernel 1: L1-normalize each query row (one block per row).
// ---------------------------------------------------------------------------
__global__ __launch_bounds__(256) void db_l1norm(const float* __restrict__ q,
                                                 float* __restrict__ qn) {
  __shared__ float red[8];
  const int row = blockIdx.x;
  const int t = threadIdx.x;
  const float* src = q + (size_t)row * D_DIM;
  float s = 0.f;
#pragma unroll
  for (int i = 0; i < 3; ++i) s += fabsf(src[t + i * 256]);
#pragma unroll
  for (int off = 16; off > 0; off >>= 1) s += __shfl_down(s, off, 32);
  if ((t & 31) == 0) red[t >> 5] = s;
  __syncthreads();
  if (t == 0) {
    float tot = 0.f;
#pragma unroll
    for (int i = 0; i < 8; ++i) tot += red[i];
    red[0] = 1.0f / fmaxf(tot, 1e-12f);
  }
  __syncthreads();
  const float inv = red[0];
  float* dst = qn + (size_t)row * D_DIM;
#pragma unroll
  for (int i = 0; i < 3; ++i) dst[t + i * 256] = src[t + i * 256] * inv;
}

// 4-slot descending insertion (strict > keeps earlier/lower index on ties).
__device__ __forceinline__ void top4_insert(float cv, int ci,
                                            float& tv0, float& tv1, float& tv2, float& tv3,
                                            int& ti0, int& ti1, int& ti2, int& ti3) {
  if (cv > tv3) {
    if (cv > tv2) {
      tv3 = tv2; ti3 = ti2;
      if (cv > tv1) {
        tv2 = tv1; ti2 = ti1;
        if (cv > tv0) { tv1 = tv0; ti1 = ti0; tv0 = cv; ti0 = ci; }
        else          { tv1 = cv;  ti1 = ci; }
      } else { tv2 = cv; ti2 = ci; }
    } else { tv3 = cv; ti3 = ci; }
  }
}

// ---------------------------------------------------------------------------
// TDM: DMA one [KTILE x NTILE] f32 slab (row stride N elements) into LDS,
// padded to BSTRIDE dwords per row via the descriptor pad feature.
// D# per CDNA5 ISA 8.3/8.4: group0 = {flags, lds_addr, global_addr, type},
// group1 = {mode/pad, dims, tile dims, strides}; groups 2/3 zero (2D tile).
// ---------------------------------------------------------------------------
__device__ __forceinline__ void tdm_load_slab(unsigned int ldsAddr,
                                              const float* gsrc, int N) {
  const unsigned long long ga = (unsigned long long)(uintptr_t)gsrc;
  u32x4 g0;
  g0[0] = 1u;                                              // count=1 valid user D#
  g0[1] = ldsAddr;                                         // lds_addr (bytes)
  g0[2] = (unsigned int)(ga & 0xFFFFFFFFull);              // global_addr[31:0]
  g0[3] = (unsigned int)((ga >> 32) & 0x01FFFFFFull)       // global_addr[56:32]
          | (2u << 30);                                    // type=2 (image)
  i32x8 g1;
  g1[0] = (int)((2u << 16)      // data_size = 4 bytes
              | (1u << 20)      // pad_enable
              | (5u << 22)      // pad_interval: 64 dwords
              | (7u << 25));    // pad_amount: 8 dwords
  g1[1] = (int)(((unsigned int)N & 0xFFFFu) << 16);                    // tensor_dim0[15:0]
  g1[2] = (int)((((unsigned int)N >> 16) & 0xFFFFu)                    // tensor_dim0[31:16]
              | ((unsigned int)D_DIM << 16));                          // tensor_dim1[15:0]
  g1[3] = (int)((unsigned int)NTILE << 16);                // dim1 hi=0 | tile_dim0=64
  g1[4] = (int)(unsigned int)KTILE;                        // tile_dim1=32, tile_dim2=0
  g1[5] = (int)N;                                          // tensor_dim0_stride[31:0]
  g1[6] = 0;                                               // stride hi / dim1_stride lo
  g1[7] = 0;
  i32x4 g2 = {0, 0, 0, 0};
  i32x4 g3 = {0, 0, 0, 0};
#if defined(__clang_major__) && (__clang_major__ >= 23)
  i32x8 g4 = {0, 0, 0, 0, 0, 0, 0, 0};
  __builtin_amdgcn_tensor_load_to_lds(g0, g1, g2, g3, g4, 0);
#else
  __builtin_amdgcn_tensor_load_to_lds(g0, g1, g2, g3, 0);
#endif
}

// ---------------------------------------------------------------------------
// Kernel 2: fused GEMM (WMMA f32 16x16x4) + mask + per-block running top-4.
//   Block = 256 threads = 8 waves; WG computes [256 x NTILE] per chunk.
//   Wave w owns rows [32w, 32w+32): two 16-row WMMA tiles x four 16-col tiles.
//   Wave 0 drives a double-buffered TDM pipeline for the embedding slabs.
// ---------------------------------------------------------------------------
__global__ __launch_bounds__(256) void db_simtopk(
    const float* __restrict__ qn, const float* __restrict__ emb,
    const int* __restrict__ startp, const int* __restrict__ endp,
    float* __restrict__ pvals, int* __restrict__ pidx,
    int nChunks, int N) {
  __shared__ float bT[2][KTILE * BSTRIDE];  // 2 x 9216 B double-buffered slabs
  __shared__ float sT[Q_DIM * SSTRIDE];     // 34816 B, 32-col sim half-tile

  const int t    = threadIdx.x;
  const int wave = t >> 5;
  const int lane = t & 31;
  const int lh   = lane >> 4;   // half-wave select (K offset pairs)
  const int l16  = lane & 15;

  const int maskStart = *startp;
  const int maskEnd   = *endp;

  // A fragment row pointers (ISA 7.12.2: lane = M row, lh picks K pair)
  const float* aRow0 = qn + (size_t)(wave * 32 + l16) * D_DIM;
  const float* aRow1 = aRow0 + (size_t)16 * D_DIM;

  const unsigned int bTaddr0 = (unsigned int)(uintptr_t)(&bT[0][0]);
  const unsigned int bTaddr1 = (unsigned int)(uintptr_t)(&bT[1][0]);

  const float NEG_INF = -__builtin_inff();
  float tv0 = NEG_INF, tv1 = NEG_INF, tv2 = NEG_INF, tv3 = NEG_INF;
  int   ti0 = -1, ti1 = -1, ti2 = -1, ti3 = -1;

  const int nSlabs = D_DIM / KTILE;  // 24

  for (int chunk = blockIdx.x; chunk < nChunks; chunk += gridDim.x) {
    const int n0 = chunk * NTILE;

    v8f acc[2][4];
#pragma unroll
    for (int i = 0; i < 2; ++i)
#pragma unroll
      for (int j = 0; j < 4; ++j)
#pragma unroll
        for (int e = 0; e < 8; ++e) acc[i][j][e] = 0.0f;

    // ---- prologue: all prior readers of bT done; kick slab 0 into buf 0
    __syncthreads();
    if (wave == 0) {
      tdm_load_slab(bTaddr0, emb + (size_t)n0, N);
    }

    for (int s = 0; s < nSlabs; ++s) {
      const int buf = s & 1;
      if (wave == 0) {
        __builtin_amdgcn_s_wait_tensorcnt(0);  // slab s landed in LDS
      }
      // Barrier: (a) publishes slab s to all waves, (b) proves every wave
      // finished computing slab s-1, so its buffer is safe to overwrite.
      __syncthreads();
      if (wave == 0 && (s + 1) < nSlabs) {
        tdm_load_slab(buf ? bTaddr0 : bTaddr1,
                      emb + (size_t)(s + 1) * KTILE * (size_t)N + (size_t)n0, N);
      }

      const int k0 = s * KTILE;
      const float* bSlab = &bT[buf][0];
      // ---- 8 x (K=4) WMMA steps over this slab
#pragma unroll
      for (int ks = 0; ks < KTILE; ks += 4) {
        const int ka = k0 + ks + lh * 2;
        const v2f a0 = *(const v2f*)(aRow0 + ka);
        const v2f a1 = *(const v2f*)(aRow1 + ka);
        const int kb = ks + lh * 2;
        v2f b[4];
#pragma unroll
        for (int nt = 0; nt < 4; ++nt) {
          b[nt][0] = bSlab[(kb + 0) * BSTRIDE + nt * 16 + l16];
          b[nt][1] = bSlab[(kb + 1) * BSTRIDE + nt * 16 + l16];
        }
#pragma unroll
        for (int nt = 0; nt < 4; ++nt) {
          acc[0][nt] = __builtin_amdgcn_wmma_f32_16x16x4_f32(
              false, a0, false, b[nt], (short)0, acc[0][nt], false, false);
          acc[1][nt] = __builtin_amdgcn_wmma_f32_16x16x4_f32(
              false, a1, false, b[nt], (short)0, acc[1][nt], false, false);
        }
      }
    }

    // ---- spill C in two 32-column halves, scan rows for top-4
#pragma unroll
    for (int h = 0; h < 2; ++h) {
      __syncthreads();  // previous half's readers (or last slab's compute) done
#pragma unroll
      for (int mt = 0; mt < 2; ++mt) {
#pragma unroll
        for (int nn = 0; nn < 2; ++nn) {
          const int nt = 2 * h + nn;
          float* dst = sT + (wave * 32 + mt * 16 + lh * 8) * SSTRIDE + nn * 16 + l16;
#pragma unroll
          for (int j = 0; j < 8; ++j) dst[j * SSTRIDE] = acc[mt][nt][j];
        }
      }
      __syncthreads();
      // thread t owns query row t
#pragma unroll 4
      for (int c = 0; c < 32; ++c) {
        const int col = n0 + h * 32 + c;
        const float cv = sT[t * SSTRIDE + c];
        const bool ok = (col < N) && !((col >= maskStart) && (col < maskEnd));
        if (ok) top4_insert(cv, col, tv0, tv1, tv2, tv3, ti0, ti1, ti2, ti3);
      }
    }
  }

  // ---- per-block partial top-4 for query row t
  const size_t base = ((size_t)blockIdx.x * Q_DIM + t) * 4;
  pvals[base + 0] = tv0; pvals[base + 1] = tv1; pvals[base + 2] = tv2; pvals[base + 3] = tv3;
  pidx[base + 0] = ti0;  pidx[base + 1] = ti1;  pidx[base + 2] = ti2;  pidx[base + 3] = ti3;
}

// ---------------------------------------------------------------------------
// Kernel 3: merge per-block partial top-4 lists; one thread per query row.
// ---------------------------------------------------------------------------
__global__ __launch_bounds__(256) void db_merge(const float* __restrict__ pvals,
                                                const int* __restrict__ pidx,
                                                float* __restrict__ out, int G) {
  const int t = threadIdx.x;
  const float NEG_INF = -__builtin_inff();
  float tv0 = NEG_INF, tv1 = NEG_INF, tv2 = NEG_INF, tv3 = NEG_INF;
  int   ti0 = -1, ti1 = -1, ti2 = -1, ti3 = -1;
  for (int g = 0; g < G; ++g) {
    const size_t base = ((size_t)g * Q_DIM + t) * 4;
#pragma unroll
    for (int j = 0; j < 4; ++j) {
      top4_insert(pvals[base + j], pidx[base + j],
                  tv0, tv1, tv2, tv3, ti0, ti1, ti2, ti3);
    }
  }
  out[t * 4 + 0] = tv0; out[t * 4 + 1] = tv1; out[t * 4 + 2] = tv2; out[t * 4 + 3] = tv3;
  float* oid = out + Q_DIM * 4;
  oid[t * 4 + 0] = (float)ti0; oid[t * 4 + 1] = (float)ti1;
  oid[t * 4 + 2] = (float)ti2; oid[t * 4 + 3] = (float)ti3;
}

// ---------------------------------------------------------------------------
extern "C" void kernel_launch(void* const* d_in, const int* in_sizes, int n_in,
                              void* d_out, int out_size, void* d_ws, size_t ws_size,
                              hipStream_t stream) {
  const float* query = (const float*)d_in[0];
  const float* emb   = (const float*)d_in[1];
  const int* startp  = (const int*)d_in[2];
  const int* endp    = (const int*)d_in[3];

  const int Qn = in_sizes[0] / D_DIM;           // 256
  const int N  = in_sizes[1] / D_DIM;           // 400000
  const int nChunks = (N + NTILE - 1) / NTILE;  // 6250

  // workspace layout: normalized q | partial values | partial indices
  float* qn    = (float*)d_ws;
  float* pvals = qn + (size_t)Qn * D_DIM;
  int*   pidx  = (int*)(pvals + (size_t)GRID_MAIN * Qn * 4);

  db_l1norm<<<Qn, 256, 0, stream>>>(query, qn);
  db_simtopk<<<GRID_MAIN, 256, 0, stream>>>(qn, emb, startp, endp,
                                            pvals, pidx, nChunks, N);
  db_merge<<<1, 256, 0, stream>>>(pvals, pidx, (float*)d_out, GRID_MAIN);
}